// MutualInformation_70342974374473
// MI455X (gfx1250) — compile-verified
//
#include <hip/hip_runtime.h>

#define NN 4096
#define DD 128
#define NSIG 75
#define LOG2E 1.4426950408889634f

typedef __attribute__((ext_vector_type(2))) float v2f;
typedef __attribute__((ext_vector_type(8))) float v8f;

// workspace layout (floats):
#define WS_SQ    0        // [4096] row squared norms
#define WS_DIST  4096     // [4096] per-gemm-block sum of sqrt(d2)
#define WS_CNT   8192     // [4096] per-gemm-block nonzero counts
#define WS_SCAL  12288    // [0]=lo  [1]=step  [2]=m_opt (= log2e / sigma_opt^2)
#define WS_SWEEP 12296    // [2*NSIG*GX] num then den partials
#define GX 256            // blocks along x for the sigma sweep
#define SIG_PER_GROUP 15
#define NGROUP 5          // 5 * 15 = 75 sigmas

// ---------------- row squared norms: one wave per row ----------------
__global__ void __launch_bounds__(256) k_sqnorm(const float* __restrict__ x,
                                                float* __restrict__ sq) {
    int wave = threadIdx.x >> 5;
    int lane = threadIdx.x & 31;
    int row  = blockIdx.x * 8 + wave;
    const float4* p = (const float4*)(x + (size_t)row * DD);
    float4 v = p[lane];                      // 32 lanes * 4 = 128 elements
    float s = v.x*v.x + v.y*v.y + v.z*v.z + v.w*v.w;
    #pragma unroll
    for (int o = 16; o > 0; o >>= 1) s += __shfl_xor(s, o, 32);
    if (lane == 0) sq[row] = s;
}

// ---------------- WMMA f32 GEMM -> d2 tiles + distance stats ----------------
__device__ inline void tile_epilogue(const v8f& c, int rowA, int h, int j,
                                     const float* __restrict__ sq, float sj,
                                     float* __restrict__ d2out,
                                     float& lsum, float& lcnt) {
    #pragma unroll
    for (int v = 0; v < 8; ++v) {
        int i = rowA + v + 8 * h;            // C/D layout: vgpr v, half h -> M = v + 8h
        float d2 = sq[i] + sj - 2.0f * c[v];
        d2 = fmaxf(d2, 0.0f);
        d2out[(size_t)i * NN + j] = d2;
        if (d2 > 0.0f) { lsum += __builtin_amdgcn_sqrtf(d2); lcnt += 1.0f; }
    }
}

__global__ void __launch_bounds__(128) k_gemm(const float* __restrict__ x,
                                              const float* __restrict__ sq,
                                              float* __restrict__ d2out,
                                              float* __restrict__ distPart,
                                              float* __restrict__ cntPart) {
    int lane = threadIdx.x & 31;
    int wave = threadIdx.x >> 5;
    int m = lane & 15;                       // row-within-tile for A, col for B
    int h = lane >> 4;                       // half-wave selects K pair
    int rowA = blockIdx.y * 64 + wave * 16;  // this wave: 16 rows x 64 cols
    int colB = blockIdx.x * 64;

    const float* aptr = x + (size_t)(rowA + m) * DD + 2 * h;
    const float* bptr = x + (size_t)(colB + m) * DD + 2 * h;

    v8f c0 = {}, c1 = {}, c2 = {}, c3 = {};
    #pragma unroll 4
    for (int k = 0; k < DD; k += 4) {
        v2f a  = *(const v2f*)(aptr + k);
        v2f b0 = *(const v2f*)(bptr + k);
        v2f b1 = *(const v2f*)(bptr + 16 * DD + k);
        v2f b2 = *(const v2f*)(bptr + 32 * DD + k);
        v2f b3 = *(const v2f*)(bptr + 48 * DD + k);
        c0 = __builtin_amdgcn_wmma_f32_16x16x4_f32(false, a, false, b0, (short)0, c0, false, false);
        c1 = __builtin_amdgcn_wmma_f32_16x16x4_f32(false, a, false, b1, (short)0, c1, false, false);
        c2 = __builtin_amdgcn_wmma_f32_16x16x4_f32(false, a, false, b2, (short)0, c2, false, false);
        c3 = __builtin_amdgcn_wmma_f32_16x16x4_f32(false, a, false, b3, (short)0, c3, false, false);
    }

    float lsum = 0.0f, lcnt = 0.0f;
    tile_epilogue(c0, rowA, h, colB +  0 + m, sq, sq[colB +  0 + m], d2out, lsum, lcnt);
    tile_epilogue(c1, rowA, h, colB + 16 + m, sq, sq[colB + 16 + m], d2out, lsum, lcnt);
    tile_epilogue(c2, rowA, h, colB + 32 + m, sq, sq[colB + 32 + m], d2out, lsum, lcnt);
    tile_epilogue(c3, rowA, h, colB + 48 + m, sq, sq[colB + 48 + m], d2out, lsum, lcnt);

    #pragma unroll
    for (int o = 16; o > 0; o >>= 1) {
        lsum += __shfl_xor(lsum, o, 32);
        lcnt += __shfl_xor(lcnt, o, 32);
    }
    __shared__ float sS[4], sC[4];
    if (lane == 0) { sS[wave] = lsum; sC[wave] = lcnt; }
    __syncthreads();
    if (threadIdx.x == 0) {
        float S = sS[0] + sS[1] + sS[2] + sS[3];
        float C = sC[0] + sC[1] + sC[2] + sC[3];
        int bid = blockIdx.y * gridDim.x + blockIdx.x;
        distPart[bid] = S;
        cntPart[bid]  = C;
    }
}

// ---------------- reduce stats -> sigma grid parameters ----------------
__global__ void __launch_bounds__(256) k_mean(const float* __restrict__ distPart,
                                              const float* __restrict__ cntPart,
                                              float* __restrict__ scal) {
    float s = 0.0f, c = 0.0f;
    for (int i = threadIdx.x; i < 4096; i += 256) { s += distPart[i]; c += cntPart[i]; }
    #pragma unroll
    for (int o = 16; o > 0; o >>= 1) { s += __shfl_xor(s, o, 32); c += __shfl_xor(c, o, 32); }
    __shared__ float sS[8], sC[8];
    int lane = threadIdx.x & 31, wave = threadIdx.x >> 5;
    if (lane == 0) { sS[wave] = s; sC[wave] = c; }
    __syncthreads();
    if (threadIdx.x == 0) {
        float S = 0.0f, C = 0.0f;
        #pragma unroll
        for (int w = 0; w < 8; ++w) { S += sS[w]; C += sC[w]; }
        float mean = S / C;
        float lo = 0.1f * mean;
        float hi = 10.0f * mean;
        scal[0] = lo;
        scal[1] = (hi - lo) / (float)NSIG;
    }
}

// ---------------- sigma sweep: num/den partials, 15 sigmas per pass ----------------
__global__ void __launch_bounds__(256) k_sweep(const float* __restrict__ d2,
                                               const float* __restrict__ L,
                                               const float* __restrict__ scal,
                                               float* __restrict__ part) {
    int g = blockIdx.y;                      // sigma group 0..4
    float lo = scal[0], step = scal[1];
    float mcoef[SIG_PER_GROUP];
    #pragma unroll
    for (int j = 0; j < SIG_PER_GROUP; ++j) {
        float sig = lo + step * (float)(g * SIG_PER_GROUP + j);
        mcoef[j] = LOG2E / (sig * sig);      // K = exp2(-d2 * mcoef)
    }
    float num[SIG_PER_GROUP], den[SIG_PER_GROUP];
    #pragma unroll
    for (int j = 0; j < SIG_PER_GROUP; ++j) { num[j] = 0.0f; den[j] = 0.0f; }

    size_t total  = (size_t)NN * NN;
    size_t idx    = (size_t)blockIdx.x * 256 + threadIdx.x;
    size_t stride = (size_t)GX * 256;
    for (; idx < total; idx += stride) {
        float d = d2[idx];
        float l = L[idx];
        if (idx + stride < total) {          // gfx1250 global_prefetch path
            __builtin_prefetch(&d2[idx + stride], 0, 1);
            __builtin_prefetch(&L[idx + stride], 0, 1);
        }
        #pragma unroll
        for (int j = 0; j < SIG_PER_GROUP; ++j) {
            float K = __builtin_amdgcn_exp2f(-d * mcoef[j]);
            num[j] = fmaf(K, l, num[j]);
            den[j] = fmaf(K, K, den[j]);
        }
    }

    __shared__ float sN[8], sD[8];
    int lane = threadIdx.x & 31, wave = threadIdx.x >> 5;
    for (int j = 0; j < SIG_PER_GROUP; ++j) {
        float n = num[j], e = den[j];
        #pragma unroll
        for (int o = 16; o > 0; o >>= 1) { n += __shfl_xor(n, o, 32); e += __shfl_xor(e, o, 32); }
        if (lane == 0) { sN[wave] = n; sD[wave] = e; }
        __syncthreads();
        if (threadIdx.x == 0) {
            float tn = 0.0f, te = 0.0f;
            #pragma unroll
            for (int w = 0; w < 8; ++w) { tn += sN[w]; te += sD[w]; }
            int s = g * SIG_PER_GROUP + j;
            part[(size_t)s * GX + blockIdx.x]          = tn;
            part[(size_t)(NSIG + s) * GX + blockIdx.x] = te;
        }
        __syncthreads();
    }
}

// ---------------- argmax over sigmas (first max, like reference) ----------------
__global__ void __launch_bounds__(128) k_pick(const float* __restrict__ part,
                                              float* __restrict__ scal) {
    __shared__ float ratio[NSIG];
    int s = threadIdx.x;
    if (s < NSIG) {
        float num = 0.0f, den = 0.0f;
        for (int b = 0; b < GX; ++b) {
            num += part[(size_t)s * GX + b];
            den += part[(size_t)(NSIG + s) * GX + b];
        }
        ratio[s] = num / __builtin_amdgcn_sqrtf(den);   // ||L|| factor cancels in argmax
    }
    __syncthreads();
    if (threadIdx.x == 0) {
        int best = 0; float bv = ratio[0];
        for (int i = 1; i < NSIG; ++i)
            if (ratio[i] > bv) { bv = ratio[i]; best = i; }   // strict > keeps first max
        float sig = scal[0] + scal[1] * (float)best;
        scal[2] = LOG2E / (sig * sig);
    }
}

// ---------------- A = exp(-d2/sigma_opt^2) / n, in place over d_out ----------------
__global__ void __launch_bounds__(256) k_final(float* __restrict__ out,
                                               const float* __restrict__ scal) {
    float m = scal[2];
    size_t total  = (size_t)NN * NN;
    size_t stride = (size_t)gridDim.x * 256;
    for (size_t idx = (size_t)blockIdx.x * 256 + threadIdx.x; idx < total; idx += stride) {
        float d = out[idx];
        out[idx] = __builtin_amdgcn_exp2f(-d * m) * (1.0f / (float)NN);
    }
}

extern "C" void kernel_launch(void* const* d_in, const int* in_sizes, int n_in,
                              void* d_out, int out_size, void* d_ws, size_t ws_size,
                              hipStream_t stream) {
    const float* x = (const float*)d_in[0];        // [4096,128]
    const float* L = (const float*)d_in[1];        // [4096,4096]
    float* out = (float*)d_out;                    // [4096,4096]
    float* ws  = (float*)d_ws;

    float* sq       = ws + WS_SQ;
    float* distPart = ws + WS_DIST;
    float* cntPart  = ws + WS_CNT;
    float* scal     = ws + WS_SCAL;
    float* sweep    = ws + WS_SWEEP;

    k_sqnorm<<<NN / 8, 256, 0, stream>>>(x, sq);

    dim3 g2(NN / 64, NN / 64);
    k_gemm<<<g2, 128, 0, stream>>>(x, sq, out, distPart, cntPart);

    k_mean<<<1, 256, 0, stream>>>(distPart, cntPart, scal);

    dim3 g3(GX, NGROUP);
    k_sweep<<<g3, 256, 0, stream>>>(out, L, scal, sweep);

    k_pick<<<1, 128, 0, stream>>>(sweep, scal);

    k_final<<<2048, 256, 0, stream>>>(out, scal);
}